// BiAttentionSeq2Span_34866544509181
// MI455X (gfx1250) — compile-verified
//
#include <hip/hip_runtime.h>
#include <math.h>

// BiDAF co-attention, b=8, L=4096, d=1024, M=2. Memory-bound streaming problem:
// out = concat([input, o1, input*o1, o2*o1]) -> 537MB write, 134MB input read x2.
// K1 uses V_WMMA_F32_16X16X4_F32 for att = [input]x[dot_scale*mem | w_input].

#define B_   8
#define L_   4096
#define D_   1024

typedef float v2f __attribute__((ext_vector_type(2)));
typedef float v4f __attribute__((ext_vector_type(4)));
typedef float v8f __attribute__((ext_vector_type(8)));

// ---------------- K0: zero o2 accumulator, compute memory_dot ----------------
__global__ __launch_bounds__(256) void k0_init(const float* __restrict__ memory,
                                               const float* __restrict__ w_memory,
                                               float* __restrict__ ws_o2,
                                               float* __restrict__ ws_md) {
  const int b = blockIdx.x;
  const int tid = threadIdx.x;
  __shared__ float r0[256];
  __shared__ float r1[256];

  for (int i = tid; i < D_; i += 256) ws_o2[b * D_ + i] = 0.0f;

  float p0 = 0.f, p1 = 0.f;
  const float* m0 = memory + (size_t)b * 2 * D_;
  const float* m1 = m0 + D_;
  for (int k = tid; k < D_; k += 256) {
    float wm = w_memory[k];
    p0 += wm * m0[k];
    p1 += wm * m1[k];
  }
  r0[tid] = p0; r1[tid] = p1;
  __syncthreads();
  for (int off = 128; off > 0; off >>= 1) {
    if (tid < off) { r0[tid] += r0[tid + off]; r1[tid] += r1[tid + off]; }
    __syncthreads();
  }
  if (tid == 0) { ws_md[b * 2 + 0] = r0[0]; ws_md[b * 2 + 1] = r1[0]; }
}

// ---------------- K1: att via WMMA f32 16x16x4; emit w1 and s ----------------
// Block: 256 threads (8 waves), handles one batch and 128 rows (16 rows/wave).
// B-matrix (d x 16): col0 = dot_scale*mem0, col1 = dot_scale*mem1, col2 = w_input,
// cols 3..15 = 0 (all mapped to a zeroed 4th LDS row -> branch-free ds_load_b64,
// no EXEC save/restore around the WMMA).  C cols: c0=cross0, c1=cross1, c2=i_dot.
#define TILE_ROWS 128
#define CHUNK_D   64
#define A_STRIDE  68          // 64 + 4 pad: 16B-aligned rows, conflict-free A reads
#define B_STRIDE  (D_ + 2)    // bank = (2n + k) % 64 -> 4 active rows on distinct banks

__global__ __launch_bounds__(256) void k1_att(const float* __restrict__ input,
                                              const float* __restrict__ memory,
                                              const float* __restrict__ w_input,
                                              const float* __restrict__ dot_scale,
                                              const float* __restrict__ ws_md,
                                              float* __restrict__ ws_w1,
                                              float* __restrict__ ws_s) {
  __shared__ float Atile[TILE_ROWS * A_STRIDE];   // 34816 B
  __shared__ float Bmat[4 * B_STRIDE];            // 16416 B (row 3 = zeros)

  const int bid    = blockIdx.x;
  const int b      = bid >> 5;          // 8 batches
  const int stripe = bid & 31;          // 32 stripes of 128 rows
  const int l0     = stripe * TILE_ROWS;
  const int tid    = threadIdx.x;
  const int lane   = tid & 31;
  const int wave   = tid >> 5;
  const int half   = lane >> 4;
  const int rowBase = wave * 16;

  const float* mem0 = memory + (size_t)b * 2 * D_;
  const float* mem1 = mem0 + D_;

  // Precompute folded B matrix columns into LDS (row 3 zeroed).
  for (int k = tid; k < D_; k += 256) {
    float ds = dot_scale[k];
    Bmat[0 * B_STRIDE + k] = ds * mem0[k];
    Bmat[1 * B_STRIDE + k] = ds * mem1[k];
    Bmat[2 * B_STRIDE + k] = w_input[k];
    Bmat[3 * B_STRIDE + k] = 0.0f;
  }
  __syncthreads();

  const float md0 = ws_md[b * 2 + 0];
  const float md1 = ws_md[b * 2 + 1];

  v8f c = {};

  const int n  = lane & 15;           // B column / C column index for this lane
  const int nb = (n < 3) ? n : 3;     // lanes 3..15 read the zero row
  const int r  = lane & 15;           // A row-in-tile for this lane

  for (int kc = 0; kc < D_ / CHUNK_D; ++kc) {
    const int d0 = kc * CHUNK_D;
    // Coalesced global -> LDS staging: 128 rows x 64 cols (float4 per thread).
    {
      const int col4 = tid & 15;          // float4 column 0..15
      const int rsub = tid >> 4;          // 0..15
      #pragma unroll
      for (int rr = 0; rr < 8; ++rr) {
        const int row = rr * 16 + rsub;
        v4f v = *(const v4f*)&input[((size_t)(b * L_ + l0 + row)) * D_ + d0 + col4 * 4];
        *(v4f*)&Atile[row * A_STRIDE + col4 * 4] = v;
      }
    }
    __syncthreads();

    // 16 WMMA ops of K=4 over this 64-wide chunk (branch-free operand loads).
    #pragma unroll
    for (int i = 0; i < 16; ++i) {
      const int klocal = 4 * i + 2 * half;          // k pair for this lane
      v2f a  = *(const v2f*)&Atile[(rowBase + r) * A_STRIDE + klocal];
      v2f bv = *(const v2f*)&Bmat[nb * B_STRIDE + d0 + klocal];
      c = __builtin_amdgcn_wmma_f32_16x16x4_f32(
              /*neg_a=*/false, a, /*neg_b=*/false, bv,
              /*c_mod=*/(short)0, c, /*reuse_a=*/false, /*reuse_b=*/false);
    }
    __syncthreads();
  }

  // Extract att columns 0..2 per row; emit weight_one and s = max_m att.
  #pragma unroll
  for (int v = 0; v < 8; ++v) {
    float cv = c[v];
    float c0 = __shfl(cv, half * 16 + 0, 32);
    float c1 = __shfl(cv, half * 16 + 1, 32);
    float c2 = __shfl(cv, half * 16 + 2, 32);
    if ((lane & 15) == 0) {
      const int row = l0 + rowBase + v + 8 * half;
      float a0 = c0 + c2 + md0;
      float a1 = c1 + c2 + md1;
      float s  = fmaxf(a0, a1);
      float e0 = __expf(a0 - s);
      float e1 = __expf(a1 - s);
      float inv = 1.0f / (e0 + e1);
      v2f w1; w1.x = e0 * inv; w1.y = e1 * inv;
      *(v2f*)&ws_w1[(size_t)(b * L_ + row) * 2] = w1;
      ws_s[b * L_ + row] = s;
    }
  }
}

// ---------------- K2: per-batch softmax over L (weight_two) ----------------
__global__ __launch_bounds__(1024) void k2_softmax(const float* __restrict__ ws_s,
                                                   float* __restrict__ ws_w2) {
  const int b = blockIdx.x;
  const int tid = threadIdx.x;
  __shared__ float red[1024];

  float sv[4];
  float mx = -INFINITY;
  #pragma unroll
  for (int i = 0; i < 4; ++i) {
    sv[i] = ws_s[b * L_ + tid + i * 1024];
    mx = fmaxf(mx, sv[i]);
  }
  red[tid] = mx;
  __syncthreads();
  for (int off = 512; off > 0; off >>= 1) {
    if (tid < off) red[tid] = fmaxf(red[tid], red[tid + off]);
    __syncthreads();
  }
  mx = red[0];
  __syncthreads();

  float sm = 0.f;
  #pragma unroll
  for (int i = 0; i < 4; ++i) sm += __expf(sv[i] - mx);
  red[tid] = sm;
  __syncthreads();
  for (int off = 512; off > 0; off >>= 1) {
    if (tid < off) red[tid] += red[tid + off];
    __syncthreads();
  }
  const float invZ = 1.0f / red[0];
  #pragma unroll
  for (int i = 0; i < 4; ++i)
    ws_w2[b * L_ + tid + i * 1024] = __expf(sv[i] - mx) * invZ;
}

// ------- K3: stream input, write out quarters 0..2, accumulate output_two -------
#define K3_ROWS 64
__global__ __launch_bounds__(256) void k3_main(const float* __restrict__ input,
                                               const float* __restrict__ memory,
                                               const float* __restrict__ ws_w1,
                                               const float* __restrict__ ws_w2,
                                               float* __restrict__ out,
                                               float* __restrict__ ws_o2) {
  const int b      = blockIdx.x >> 6;       // 64 stripes per batch
  const int stripe = blockIdx.x & 63;
  const int l0     = stripe * K3_ROWS;
  const int t      = threadIdx.x;           // float4 column 0..255

  const float* mb = memory + (size_t)b * 2 * D_;
  v4f m0 = *(const v4f*)&mb[4 * t];
  v4f m1 = *(const v4f*)&mb[D_ + 4 * t];
  v4f acc = {};

  for (int rIt = 0; rIt < K3_ROWS; ++rIt) {
    const int l = l0 + rIt;
    const size_t rowIn  = (size_t)(b * L_ + l) * D_;
    const size_t rowOut = (size_t)(b * L_ + l) * (4 * D_);
    v4f x = *(const v4f*)&input[rowIn + 4 * t];
    v2f w1 = *(const v2f*)&ws_w1[(size_t)(b * L_ + l) * 2];
    float w2l = ws_w2[b * L_ + l];
    v4f o1 = w1.x * m0 + w1.y * m1;
    *(v4f*)&out[rowOut + 0 * D_ + 4 * t] = x;
    *(v4f*)&out[rowOut + 1 * D_ + 4 * t] = o1;
    *(v4f*)&out[rowOut + 2 * D_ + 4 * t] = x * o1;
    acc += w2l * x;
  }
  float* o2p = ws_o2 + (size_t)b * D_ + 4 * t;
  atomicAdd(o2p + 0, acc.x);
  atomicAdd(o2p + 1, acc.y);
  atomicAdd(o2p + 2, acc.z);
  atomicAdd(o2p + 3, acc.w);
}

// ---------------- K4: out quarter 3 = output_two * output_one ----------------
__global__ __launch_bounds__(256) void k4_tail(const float* __restrict__ memory,
                                               const float* __restrict__ ws_w1,
                                               const float* __restrict__ ws_o2,
                                               float* __restrict__ out) {
  const int b      = blockIdx.x >> 6;
  const int stripe = blockIdx.x & 63;
  const int l0     = stripe * K3_ROWS;
  const int t      = threadIdx.x;

  const float* mb = memory + (size_t)b * 2 * D_;
  v4f m0  = *(const v4f*)&mb[4 * t];
  v4f m1  = *(const v4f*)&mb[D_ + 4 * t];
  v4f o2v = *(const v4f*)&ws_o2[(size_t)b * D_ + 4 * t];

  for (int rIt = 0; rIt < K3_ROWS; ++rIt) {
    const int l = l0 + rIt;
    v2f w1 = *(const v2f*)&ws_w1[(size_t)(b * L_ + l) * 2];
    v4f o1 = w1.x * m0 + w1.y * m1;
    const size_t rowOut = (size_t)(b * L_ + l) * (4 * D_);
    *(v4f*)&out[rowOut + 3 * D_ + 4 * t] = o2v * o1;
  }
}

extern "C" void kernel_launch(void* const* d_in, const int* in_sizes, int n_in,
                              void* d_out, int out_size, void* d_ws, size_t ws_size,
                              hipStream_t stream) {
  const float* input     = (const float*)d_in[0];
  const float* memory    = (const float*)d_in[1];
  const float* w_input   = (const float*)d_in[2];
  const float* w_memory  = (const float*)d_in[3];
  const float* dot_scale = (const float*)d_in[4];
  float* out = (float*)d_out;
  float* ws  = (float*)d_ws;

  // Workspace layout (floats): w1[B*L*2] | s[B*L] | w2[B*L] | o2[B*D] | md[B*2]
  float* ws_w1 = ws;                       // 65536
  float* ws_s  = ws + 65536;               // 32768
  float* ws_w2 = ws + 98304;               // 32768
  float* ws_o2 = ws + 131072;              // 8192
  float* ws_md = ws + 139264;              // 16   (total ~557 KB)

  k0_init   <<<B_,        256,  0, stream>>>(memory, w_memory, ws_o2, ws_md);
  k1_att    <<<B_ * 32,   256,  0, stream>>>(input, memory, w_input, dot_scale,
                                             ws_md, ws_w1, ws_s);
  k2_softmax<<<B_,        1024, 0, stream>>>(ws_s, ws_w2);
  k3_main   <<<B_ * 64,   256,  0, stream>>>(input, memory, ws_w1, ws_w2, out, ws_o2);
  k4_tail   <<<B_ * 64,   256,  0, stream>>>(memory, ws_w1, ws_o2, out);
}